// GCN_50912542326918
// MI455X (gfx1250) — compile-verified
//
#include <hip/hip_runtime.h>

typedef __attribute__((ext_vector_type(2))) float v2f;
typedef __attribute__((ext_vector_type(4))) float v4f;
typedef __attribute__((ext_vector_type(8))) float v8f;

#define FEAT 64
#define WAVES_PER_BLOCK 8
#define ROWS_PER_WAVE 16
#define XPITCH 65  // 64 + 1 pad to avoid LDS bank conflicts on column reads

// -----------------------------------------------------------------------------
// Kernel 1: fused dual GEMM via V_WMMA_F32_16X16X4_F32
//   t   = x @ W          (written to workspace; scattered along edges later)
//   out = x @ S + bias   (initializes the output; scatter adds on top)
// Each wave owns a 16-row tile and produces a 16x128 result ([t | out]).
// Combined B = [W | S] (64 x 128) staged once per block in LDS.
// -----------------------------------------------------------------------------
__global__ __launch_bounds__(256) void gcn_gemm(const float* __restrict__ x,
                                                const float* __restrict__ W,
                                                const float* __restrict__ S,
                                                const float* __restrict__ bias,
                                                float* __restrict__ t,
                                                float* __restrict__ out,
                                                int n_nodes)
{
    __shared__ float ldsB[64 * 128];                       // [k][j]: j<64 -> W, j>=64 -> S
    __shared__ float ldsX[WAVES_PER_BLOCK][ROWS_PER_WAVE * XPITCH];

    const int wave = threadIdx.x >> 5;
    const int lane = threadIdx.x & 31;
    const int tile = blockIdx.x * WAVES_PER_BLOCK + wave;
    const int rowBase = tile * ROWS_PER_WAVE;
    const bool active = (rowBase < n_nodes);

    // --- cooperative load of combined B into LDS (all 256 threads) ---
    for (int i = threadIdx.x; i < 64 * 128; i += 256) {
        const int k = i >> 7;
        const int c = i & 127;
        ldsB[i] = (c < 64) ? W[k * 64 + c] : S[k * 64 + (c - 64)];
    }

    // --- per-wave coalesced staging of the 16x64 x tile (row-clamped) ---
    if (active) {
        float* lx = &ldsX[wave][0];
        const size_t base = (size_t)rowBase * FEAT;
        #pragma unroll
        for (int i = lane; i < ROWS_PER_WAVE * FEAT; i += 32) {
            const int rr = i >> 6;
            const int cc = i & 63;
            const float v = (rowBase + rr < n_nodes) ? x[base + i] : 0.0f;
            lx[rr * XPITCH + cc] = v;
        }
    }

    __syncthreads();
    if (!active) return;

    const int m  = lane & 15;          // A: row in tile; B/D: column within 16-wide tile
    const int kh = (lane >> 4) * 2;    // K contribution of upper/lower lane half
    const float* lx = &ldsX[wave][0];

    v8f acc[8];
    #pragma unroll
    for (int j = 0; j < 8; ++j)
        acc[j] = (v8f){0.f, 0.f, 0.f, 0.f, 0.f, 0.f, 0.f, 0.f};

    for (int k = 0; k < FEAT; k += 4) {
        // A fragment: A[M = m][K = k + kh + v]
        v2f a;
        a.x = lx[m * XPITCH + k + kh + 0];
        a.y = lx[m * XPITCH + k + kh + 1];
        #pragma unroll
        for (int j = 0; j < 8; ++j) {
            // B fragment: B[K = k + kh + v][N = j*16 + m]
            v2f b;
            b.x = ldsB[(k + kh + 0) * 128 + j * 16 + m];
            b.y = ldsB[(k + kh + 1) * 128 + j * 16 + m];
            acc[j] = __builtin_amdgcn_wmma_f32_16x16x4_f32(
                false, a, false, b, (short)0, acc[j], false, false);
        }
    }

    // D layout: element (VGPR r, lane L) = D[M = r + (L>=16)*8][N = m]
    const int rowOff = (lane >> 4) * 8;
    #pragma unroll
    for (int j = 0; j < 4; ++j) {                  // t = x @ W
        float* dst = t + (size_t)(rowBase + rowOff) * FEAT + j * 16 + m;
        #pragma unroll
        for (int r = 0; r < 8; ++r)
            if (rowBase + rowOff + r < n_nodes)
                dst[(size_t)r * FEAT] = acc[j][r];
    }
    #pragma unroll
    for (int j = 0; j < 4; ++j) {                  // out = x @ S + bias
        const float bv = bias[j * 16 + m];
        float* dst = out + (size_t)(rowBase + rowOff) * FEAT + j * 16 + m;
        #pragma unroll
        for (int r = 0; r < 8; ++r)
            if (rowBase + rowOff + r < n_nodes)
                dst[(size_t)r * FEAT] = acc[4 + j][r] + bv;
    }
}

// -----------------------------------------------------------------------------
// Kernel 2: edge scatter  out[row] += t[col]
// 16 lanes per edge, float4 gather (global_load_b128) + 4 f32 L2 atomics each.
// -----------------------------------------------------------------------------
__global__ __launch_bounds__(256) void gcn_scatter(const long long* __restrict__ rows,
                                                   const long long* __restrict__ cols,
                                                   const float* __restrict__ t,
                                                   float* __restrict__ out,
                                                   long long n_edges)
{
    const long long gid = (long long)blockIdx.x * blockDim.x + threadIdx.x;
    const long long e = gid >> 4;
    const int q = (int)(gid & 15);           // 16 lanes/edge, 4 feats each
    if (e >= n_edges) return;

    const long long r = rows[e];
    const long long c = cols[e];

    const v4f v = *((const v4f*)(t + c * FEAT) + q);
    float* dst = out + r * FEAT + (size_t)q * 4;
    atomicAdd(dst + 0, v.x);
    atomicAdd(dst + 1, v.y);
    atomicAdd(dst + 2, v.z);
    atomicAdd(dst + 3, v.w);
}

extern "C" void kernel_launch(void* const* d_in, const int* in_sizes, int n_in,
                              void* d_out, int out_size, void* d_ws, size_t ws_size,
                              hipStream_t stream)
{
    const float*      x    = (const float*)d_in[0];
    const long long*  edge = (const long long*)d_in[1];
    const float*      W    = (const float*)d_in[2];
    const float*      S    = (const float*)d_in[3];
    const float*      bias = (const float*)d_in[4];
    float* out = (float*)d_out;
    float* t   = (float*)d_ws;   // [n_nodes, 64] fp32 scratch = 25.6 MB

    const int       n_nodes = in_sizes[0] / FEAT;        // 100000
    const long long n_edges = (long long)in_sizes[1] / 2; // 1600000

    const int rowsPerBlock = WAVES_PER_BLOCK * ROWS_PER_WAVE; // 128
    const int gemmBlocks = (n_nodes + rowsPerBlock - 1) / rowsPerBlock;
    gcn_gemm<<<gemmBlocks, 256, 0, stream>>>(x, W, S, bias, t, out, n_nodes);

    const long long scatterThreads = n_edges * 16;
    const int scatterBlocks = (int)((scatterThreads + 255) / 256);
    gcn_scatter<<<scatterBlocks, 256, 0, stream>>>(edge, edge + n_edges, t, out, n_edges);
}